// ExponentialPool_70300024701584
// MI455X (gfx1250) — compile-verified
//
#include <hip/hip_runtime.h>

// CDNA5 / gfx1250. wave32.
typedef float v2f __attribute__((ext_vector_type(2)));
typedef float v8f __attribute__((ext_vector_type(8)));

#define N_BASIS   32
#define N_ANCHOR  2048
#define N_POINTS  8192
#define NWAVES    8               // waves per block
#define NTILES    (N_ANCHOR / 32) // 64 anchor tiles

// Computes per-(anchor-tile, point-split) partial sums:
//   dst[(tile*psplit+split)*1024 + a*32 + b] = sum_j exp(-mu_b d(a,j)) f_j
__global__ __launch_bounds__(256) void expo_pool_partial(
    const float* __restrict__ f,
    const float* __restrict__ coords,        // (8192,3)
    const float* __restrict__ anchors,       // (2048,3)
    const float* __restrict__ mu,            // (32,)
    float* __restrict__ dst,
    int psplit)
{
    __shared__ __align__(16) float part[NWAVES][32][32];   // 32 KB

    const int tid     = threadIdx.x;
    const int wave    = tid >> 5;
    const int lane    = tid & 31;
    const int halfsel = lane >> 4;     // 0: lanes 0-15, 1: lanes 16-31
    const int lcol    = lane & 15;
    const int tileBase = blockIdx.x * 32;      // 32 anchors per block

    // ---- per-lane anchor (lane <-> anchor within tile) ----
    const int a = tileBase + lane;
    const float ax = anchors[3*a + 0];
    const float ay = anchors[3*a + 1];
    const float az = anchors[3*a + 2];
    const float na = ax*ax + ay*ay + az*az;    // |a|^2

    // ---- WMMA A operands ----
    // 32-bit A 16x4 layout: lanes 0-15 hold K0 (VGPR0), K1 (VGPR1);
    // lanes 16-31 hold K2 (VGPR0), K3 (VGPR1).  A row r = [ax, ay, az, |a|^2]
    const float xl = __shfl(ax, lcol),      yl = __shfl(ay, lcol);
    const float zl = __shfl(az, lcol),      nl = __shfl(na, lcol);
    const float xh = __shfl(ax, 16 + lcol), yh = __shfl(ay, 16 + lcol);
    const float zh = __shfl(az, 16 + lcol), nh = __shfl(na, 16 + lcol);
    v2f A_lo, A_hi;
    A_lo.x = halfsel ? zl : xl;  A_lo.y = halfsel ? nl : yl;   // anchors 0-15
    A_hi.x = halfsel ? zh : xh;  A_hi.y = halfsel ? nh : yh;   // anchors 16-31

    // exp(-mu d) = exp2(m2 * d) with m2 = -mu*log2(e)  (uniform -> SGPRs)
    float m2[N_BASIS];
#pragma unroll
    for (int b = 0; b < N_BASIS; ++b)
        m2[b] = mu[b] * (-1.4426950408889634f);

    float acc[N_BASIS];
#pragma unroll
    for (int b = 0; b < N_BASIS; ++b) acc[b] = 0.0f;

    float* tr = &part[wave][0][0];             // 32x16 scratch (16-col rows)
    const int ptsPerSplit = N_POINTS / psplit;
    const int waveBase = blockIdx.y * ptsPerSplit + wave * (ptsPerSplit / NWAVES);
    const int chunks   = ptsPerSplit / (NWAVES * 16);

    for (int c = 0; c < chunks; ++c) {
        const int base = waveBase + c * 16;
        const int j    = base + lcol;          // both half-lanes load same point
        const float cx = coords[3*j + 0];
        const float cy = coords[3*j + 1];
        const float cz = coords[3*j + 2];
        const float nc = cx*cx + cy*cy + cz*cz;   // |c|^2

        // B 4x16 (row striped across lanes within a VGPR):
        //   VGPR0: lanes0-15 = K0 row, lanes16-31 = K1 row
        //   VGPR1: lanes0-15 = K2 row, lanes16-31 = K3 row
        // B col j = [-2cx, -2cy, -2cz, 1]
        v2f B;
        B.x = halfsel ? (-2.0f * cy) : (-2.0f * cx);
        B.y = halfsel ? 1.0f         : (-2.0f * cz);

        v8f C;
#pragma unroll
        for (int i = 0; i < 8; ++i) C[i] = nc;    // seed with |c_j|^2

        // D[r,j] = |a_r|^2 - 2 a_r.c_j + |c_j|^2 = d^2
        v8f d2lo = __builtin_amdgcn_wmma_f32_16x16x4_f32(
            false, A_lo, false, B, (short)0, C, false, false);
        v8f d2hi = __builtin_amdgcn_wmma_f32_16x16x4_f32(
            false, A_hi, false, B, (short)0, C, false, false);

        // ---- transpose 32x16 d^2 tile through LDS so lane == anchor ----
#pragma unroll
        for (int v = 0; v < 8; ++v) {
            tr[(v + 8*halfsel) * 16 + lcol]      = d2lo[v];   // rows 0-15
            tr[(16 + v + 8*halfsel) * 16 + lcol] = d2hi[v];   // rows 16-31
        }
        __syncthreads();

        float d2r[16];
#pragma unroll
        for (int q = 0; q < 4; ++q) {
            float4 t = *reinterpret_cast<const float4*>(&tr[lane*16 + 4*q]);
            d2r[4*q+0] = t.x; d2r[4*q+1] = t.y;
            d2r[4*q+2] = t.z; d2r[4*q+3] = t.w;
        }

        float fr[16];
#pragma unroll
        for (int q = 0; q < 4; ++q) {
            float4 t = *reinterpret_cast<const float4*>(f + base + 4*q);
            fr[4*q+0] = t.x; fr[4*q+1] = t.y;
            fr[4*q+2] = t.z; fr[4*q+3] = t.w;
        }

        // ---- hot loop: hand-pipelined mul/exp/fmac, groups of 4 ----
        // Ordering inside the asm blocks guarantees every v_exp_f32 result is
        // consumed >=3 instructions later (ISA TRANS hazard needs >=1), so no
        // v_nop / TRANS32_DEP stalls. exp (TRANS pipe) overlaps fmac/mul (VALU).
#pragma unroll
        for (int jj = 0; jj < 16; ++jj) {
            const float d  = __builtin_amdgcn_sqrtf(fmaxf(d2r[jj], 0.0f));
            const float fj = fr[jj];
            float t0, t1, t2, t3;
            // prologue: group 0 muls+exps (v_nop shields the preceding v_sqrt)
            asm volatile(
                "v_nop\n\t"
                "v_mul_f32 %0, %4, %8\n\t"
                "v_mul_f32 %1, %5, %8\n\t"
                "v_mul_f32 %2, %6, %8\n\t"
                "v_mul_f32 %3, %7, %8\n\t"
                "v_exp_f32 %0, %0\n\t"
                "v_exp_f32 %1, %1\n\t"
                "v_exp_f32 %2, %2\n\t"
                "v_exp_f32 %3, %3"
                : "=&v"(t0), "=&v"(t1), "=&v"(t2), "=&v"(t3)
                : "s"(m2[0]), "s"(m2[1]), "s"(m2[2]), "s"(m2[3]), "v"(d));
#pragma unroll
            for (int g = 1; g < 8; ++g) {
                // fmac group g-1 (consumes exps >=3 instrs old), mul+exp group g
                asm volatile(
                    "v_fmac_f32 %0, %4, %13\n\t"
                    "v_fmac_f32 %1, %5, %13\n\t"
                    "v_fmac_f32 %2, %6, %13\n\t"
                    "v_fmac_f32 %3, %7, %13\n\t"
                    "v_mul_f32 %4, %8, %12\n\t"
                    "v_mul_f32 %5, %9, %12\n\t"
                    "v_mul_f32 %6, %10, %12\n\t"
                    "v_mul_f32 %7, %11, %12\n\t"
                    "v_exp_f32 %4, %4\n\t"
                    "v_exp_f32 %5, %5\n\t"
                    "v_exp_f32 %6, %6\n\t"
                    "v_exp_f32 %7, %7"
                    : "+v"(acc[4*(g-1)+0]), "+v"(acc[4*(g-1)+1]),
                      "+v"(acc[4*(g-1)+2]), "+v"(acc[4*(g-1)+3]),
                      "+v"(t0), "+v"(t1), "+v"(t2), "+v"(t3)
                    : "s"(m2[4*g+0]), "s"(m2[4*g+1]),
                      "s"(m2[4*g+2]), "s"(m2[4*g+3]),
                      "v"(d), "v"(fj));
            }
            // epilogue: fmac last group
            asm volatile(
                "v_fmac_f32 %0, %4, %8\n\t"
                "v_fmac_f32 %1, %5, %8\n\t"
                "v_fmac_f32 %2, %6, %8\n\t"
                "v_fmac_f32 %3, %7, %8"
                : "+v"(acc[28]), "+v"(acc[29]), "+v"(acc[30]), "+v"(acc[31])
                : "v"(t0), "v"(t1), "v"(t2), "v"(t3), "v"(fj));
        }
        __syncthreads();   // tr reused next iteration
    }

    // ---- deterministic cross-wave reduction, write partial tile ----
#pragma unroll
    for (int b = 0; b < N_BASIS; ++b)
        part[wave][lane][b] = acc[b];
    __syncthreads();

    float* dstTile = dst + (size_t)(blockIdx.x * psplit + blockIdx.y) * 1024;
    for (int e = tid; e < 32 * 32; e += 256) {
        const int aa = e >> 5;
        const int b  = e & 31;
        float s = 0.0f;
#pragma unroll
        for (int w = 0; w < NWAVES; ++w) s += part[w][aa][b];
        dstTile[e] = s;                         // e == aa*32 + b
    }
}

// out[tile*1024 + e] = (sum_p src[(tile*psplit+p)*1024 + e]) / norms[e&31]
// Safe when src == out with psplit == 1 (pure per-element in-place divide).
__global__ __launch_bounds__(256) void expo_pool_reduce(
    const float* __restrict__ src,
    const float* __restrict__ norms,
    float* __restrict__ out,
    int psplit)
{
    const int tile = blockIdx.x;
    const float* srcTile = src + (size_t)tile * psplit * 1024;
    for (int e = threadIdx.x; e < 1024; e += 256) {
        float s = 0.0f;
        for (int p = 0; p < psplit; ++p) s += srcTile[p * 1024 + e];
        out[tile * 1024 + e] = s / norms[e & 31];
    }
}

extern "C" void kernel_launch(void* const* d_in, const int* in_sizes, int n_in,
                              void* d_out, int out_size, void* d_ws, size_t ws_size,
                              hipStream_t stream) {
    const float* f       = (const float*)d_in[0];   // (8192,1)
    const float* coords  = (const float*)d_in[1];   // (8192,3)
    const float* anchors = (const float*)d_in[2];   // (2048,3)
    const float* mu      = (const float*)d_in[3];   // (32,)
    const float* norms   = (const float*)d_in[4];   // (32,)
    float* out           = (float*)d_out;           // (2048,32)

    // Pick largest point-split the workspace can hold (deterministic: ws_size
    // is fixed across calls). Partials: NTILES * psplit * 1024 floats.
    const size_t per = (size_t)NTILES * 1024 * sizeof(float);   // 256 KB
    int psplit;
    float* dst;
    if      (ws_size >= 8 * per) { psplit = 8; dst = (float*)d_ws; }
    else if (ws_size >= 4 * per) { psplit = 4; dst = (float*)d_ws; }
    else if (ws_size >= 2 * per) { psplit = 2; dst = (float*)d_ws; }
    else if (ws_size >= 1 * per) { psplit = 1; dst = (float*)d_ws; }
    else                         { psplit = 1; dst = out; }      // in-place path

    dim3 grid(NTILES, psplit);
    dim3 block(32 * NWAVES);
    expo_pool_partial<<<grid, block, 0, stream>>>(f, coords, anchors, mu, dst, psplit);
    expo_pool_reduce<<<dim3(NTILES), block, 0, stream>>>(dst, norms, out, psplit);
}